// _Interactions_45449343926354
// MI455X (gfx1250) — compile-verified
//
#include <hip/hip_runtime.h>
#include <hip/hip_bf16.h>

typedef __attribute__((ext_vector_type(2))) float v2f;
typedef __attribute__((ext_vector_type(8))) float v8f;

// ---------------------------------------------------------------------------
// Elementwise / edge kernels (bandwidth + atomic bound part of the graph conv)
// ---------------------------------------------------------------------------

__global__ void zero_kernel(float* __restrict__ p, int n) {
    int i = blockIdx.x * blockDim.x + threadIdx.x;
    if (i < n) p[i] = 0.0f;
}

__global__ void deg_kernel(const int* __restrict__ dst, const float* __restrict__ w,
                           float* __restrict__ deg, int E) {
    int e = blockIdx.x * blockDim.x + threadIdx.x;
    if (e < E) atomicAdd(&deg[dst[e]], w[e]);
}

// in-place deg -> deg^{-1/2} (0 where deg <= 0)
__global__ void dinv_kernel(float* __restrict__ deg, int N) {
    int n = blockIdx.x * blockDim.x + threadIdx.x;
    if (n < N) {
        float d = deg[n];
        deg[n] = (d > 0.0f) ? rsqrtf(d) : 0.0f;
    }
}

__global__ void norm_kernel(const int* __restrict__ src, const int* __restrict__ dst,
                            const float* __restrict__ w, const float* __restrict__ dinv,
                            float* __restrict__ norm, int E) {
    int e = blockIdx.x * blockDim.x + threadIdx.x;
    if (e < E) norm[e] = dinv[src[e]] * w[e] * dinv[dst[e]];
}

// agg[dst[e], :] += norm[e] * h[src[e], :]   (16 threads per edge, float4 each)
__global__ void edge_msg_kernel(const float* __restrict__ h, const float* __restrict__ norm,
                                const int* __restrict__ src, const int* __restrict__ dst,
                                float* __restrict__ agg, int E) {
    int t  = blockIdx.x * blockDim.x + threadIdx.x;
    int e  = t >> 4;
    int c4 = (t & 15) * 4;
    if (e >= E) return;
    float nrm = norm[e];
    const float4 hv = *reinterpret_cast<const float4*>(h + (size_t)src[e] * 64 + c4);
    float* base = agg + (size_t)dst[e] * 64 + c4;
    atomicAdd(base + 0, nrm * hv.x);
    atomicAdd(base + 1, nrm * hv.y);
    atomicAdd(base + 2, nrm * hv.z);
    atomicAdd(base + 3, nrm * hv.w);
}

// out += relu(agg + b[channel])
__global__ void residual_kernel(float* __restrict__ out, const float* __restrict__ agg,
                                const float* __restrict__ b, int total) {
    int i = blockIdx.x * blockDim.x + threadIdx.x;
    if (i < total) {
        float v = agg[i] + b[i & 63];
        out[i] += fmaxf(v, 0.0f);
    }
}

// ---------------------------------------------------------------------------
// fp32 WMMA GEMM: Out[N,64] = A[N,64] @ W[64,64] (+ bias, relu optional)
// One wave -> one 16x16 output tile, 4 waves/block cover the 64 columns,
// gridDim.x = N/16 row strips. K=64 consumed as 16 x V_WMMA_F32_16X16X4_F32.
// EXEC is all-ones (no divergence) as WMMA requires.
// ---------------------------------------------------------------------------
__global__ __launch_bounds__(128) void gemm64_wmma(const float* __restrict__ A,
                                                   const float* __restrict__ W,
                                                   const float* __restrict__ bias,
                                                   float* __restrict__ Out,
                                                   int applyBiasRelu) {
    const int lane = threadIdx.x & 31;
    const int wave = threadIdx.x >> 5;   // 0..3 -> 16-column tile
    const int r0   = blockIdx.x * 16;
    const int c0   = wave * 16;
    const int half = lane >> 4;          // 0: K even pair low, 1: K pair high
    const int lm   = lane & 15;          // M index (A) / N index (B)

    // A fragment base: lane holds A[r0+lm][k4 + 2*half + {0,1}] (contiguous f2)
    const float* arow = A + (size_t)(r0 + lm) * 64 + 2 * half;
    // B fragment base: lane holds W[k4 + 2*half + {0,1}][c0+lm]
    const float* bcol = W + (size_t)(2 * half) * 64 + c0 + lm;

    v8f acc = {};
#pragma unroll
    for (int kk = 0; kk < 16; ++kk) {
        v2f a, b;
        const float* ap = arow + kk * 4;
        a.x = ap[0];
        a.y = ap[1];
        const float* bp = bcol + (size_t)(kk * 4) * 64;
        b.x = bp[0];
        b.y = bp[64];
        acc = __builtin_amdgcn_wmma_f32_16x16x4_f32(
            /*neg_a=*/false, a, /*neg_b=*/false, b,
            /*c_mod=*/(short)0, acc, /*reuse_a=*/false, /*reuse_b=*/false);
    }

    float bv = 0.0f;
    if (applyBiasRelu) bv = bias[c0 + lm];   // uniform branch
#pragma unroll
    for (int v = 0; v < 8; ++v) {
        int m = v + 8 * half;                // C/D layout: row = vgpr + 8*half
        float val = acc[v];
        if (applyBiasRelu) val = fmaxf(val + bv, 0.0f);
        Out[(size_t)(r0 + m) * 64 + c0 + lm] = val;
    }
}

// Scalar tail for row counts not divisible by 16 (N=50000 is, so normally unused).
__global__ void gemm64_tail(const float* __restrict__ A, const float* __restrict__ W,
                            const float* __restrict__ bias, float* __restrict__ Out,
                            int rowStart, int nRows, int applyBiasRelu) {
    int idx = blockIdx.x * blockDim.x + threadIdx.x;
    if (idx >= nRows * 64) return;
    int r = rowStart + idx / 64;
    int c = idx & 63;
    float s = 0.0f;
    for (int k = 0; k < 64; ++k) s += A[(size_t)r * 64 + k] * W[(size_t)k * 64 + c];
    if (applyBiasRelu) s = fmaxf(s + bias[c], 0.0f);
    Out[(size_t)r * 64 + c] = s;
}

// ---------------------------------------------------------------------------

static inline size_t align256f(size_t v) { return (v + 255) & ~(size_t)255; }

extern "C" void kernel_launch(void* const* d_in, const int* in_sizes, int n_in,
                              void* d_out, int out_size, void* d_ws, size_t ws_size,
                              hipStream_t stream) {
    const float* x      = (const float*)d_in[0];   // [N,64]
    const int*   ei     = (const int*)d_in[1];     // [2,E] row-major: src then dst
    const float* ew     = (const float*)d_in[2];   // [E]
    /* d_in[3] edge_attr unused by the math */
    const float* lin0_w = (const float*)d_in[4];   // [64,64]
    const float* lin0_b = (const float*)d_in[5];   // [64]
    const float* conv_w = (const float*)d_in[6];   // [2,64,64]
    const float* conv_b = (const float*)d_in[7];   // [2,64]

    const int N = in_sizes[0] / 64;
    const int E = in_sizes[2];
    const int* src = ei;
    const int* dst = ei + E;
    float* out = (float*)d_out;

    // workspace carve-out (floats): deg/dinv [N], norm [E], h [N*64], agg [N*64]
    float* deg  = (float*)d_ws;
    float* norm = deg + align256f((size_t)N);
    float* h    = norm + align256f((size_t)E);
    float* agg  = h + align256f((size_t)N * 64);

    const int TPB = 256;
    const int ND  = N * 64;

    // degree + symmetric normalization
    zero_kernel<<<(N + TPB - 1) / TPB, TPB, 0, stream>>>(deg, N);
    deg_kernel <<<(E + TPB - 1) / TPB, TPB, 0, stream>>>(dst, ew, deg, E);
    dinv_kernel<<<(N + TPB - 1) / TPB, TPB, 0, stream>>>(deg, N);
    norm_kernel<<<(E + TPB - 1) / TPB, TPB, 0, stream>>>(src, dst, ew, deg, norm, E);

    const int tiles    = N / 16;
    const int tailRows = N - tiles * 16;

    // out = relu(x @ lin0_w + lin0_b)
    if (tiles > 0)
        gemm64_wmma<<<tiles, 128, 0, stream>>>(x, lin0_w, lin0_b, out, 1);
    if (tailRows > 0)
        gemm64_tail<<<(tailRows * 64 + TPB - 1) / TPB, TPB, 0, stream>>>(
            x, lin0_w, lin0_b, out, tiles * 16, tailRows, 1);

    for (int i = 0; i < 2; ++i) {
        const float* Wi = conv_w + (size_t)i * 64 * 64;
        const float* bi = conv_b + (size_t)i * 64;

        // h = out @ conv_w[i]
        if (tiles > 0)
            gemm64_wmma<<<tiles, 128, 0, stream>>>(out, Wi, nullptr, h, 0);
        if (tailRows > 0)
            gemm64_tail<<<(tailRows * 64 + TPB - 1) / TPB, TPB, 0, stream>>>(
                out, Wi, nullptr, h, tiles * 16, tailRows, 0);

        // agg = scatter_add(norm * h[src], dst)
        zero_kernel<<<(ND + TPB - 1) / TPB, TPB, 0, stream>>>(agg, ND);
        {
            long long tthreads = (long long)E * 16;
            int blocks = (int)((tthreads + TPB - 1) / TPB);
            edge_msg_kernel<<<blocks, TPB, 0, stream>>>(h, norm, src, dst, agg, E);
        }

        // out = out + relu(agg + conv_b[i])
        residual_kernel<<<(ND + TPB - 1) / TPB, TPB, 0, stream>>>(out, agg, bi, ND);
    }
}